// NeighSuperpixelAttn_16398185136396
// MI455X (gfx1250) — compile-verified
//
#include <hip/hip_runtime.h>
#include <hip/hip_bf16.h>

typedef __attribute__((ext_vector_type(2))) float v2f;
typedef __attribute__((ext_vector_type(8))) float v8f;

#define BATCH   4
#define HDIM    256
#define WDIM    256
#define CDIM    128      // K of the GEMM
#define NDIM    256      // 2*C
#define HEADS   4
#define HDSZ    32       // head dim
#define KS      7
#define M_TOTAL (BATCH * HDIM * WDIM)          // 262144 rows
#define Q_SCALE 0.17677669529663687f           // 32^-0.5

__device__ __forceinline__ int iclamp(int v, int lo, int hi) {
    return v < lo ? lo : (v > hi ? hi : v);
}
__device__ __forceinline__ int imin(int a, int b) { return a < b ? a : b; }

// ---------------------------------------------------------------------------
// Phase 1: qk = x @ W_qk + b_qk  (fp32 WMMA 16x16x4), q half pre-scaled.
// Each wave: one 16-row M tile x 128 cols (8 N tiles). 4096 blocks x 8 waves
// = 32768 wave tasks = 16384 mtiles x 2 ngroups.
// ---------------------------------------------------------------------------
__global__ __launch_bounds__(256) void qk_gemm_wmma(
    const float* __restrict__ x,     // [M_TOTAL, 128]
    const float* __restrict__ Wqk,   // [128, 256]
    const float* __restrict__ bias,  // [256]
    float* __restrict__ qk)          // [M_TOTAL, 256]
{
    const int lane  = threadIdx.x & 31;
    const int wave  = threadIdx.x >> 5;
    const int task  = blockIdx.x * 8 + wave;
    const int mt    = task >> 1;          // 16384 M tiles
    const int ng    = task & 1;           // 2 groups of 128 cols
    const int m0    = mt * 16;
    const int n0    = ng * 128;

    const int lhalf = lane >> 4;          // 0: K,K+1   1: K+2,K+3
    const int l15   = lane & 15;

    v8f acc[8] = {};                      // zero-init accumulators

    // A row pointer: lane handles row m0+l15, K offset lhalf*2
    const float* arow = x + (size_t)(m0 + l15) * CDIM + lhalf * 2;

    for (int k0 = 0; k0 < CDIM; k0 += 4) {
        // A fragment: 16x4 f32 -> 2 VGPRs (b64 load per lane)
        v2f a = *(const v2f*)(arow + k0);
        const int kb = k0 + lhalf * 2;
#pragma unroll
        for (int j = 0; j < 8; ++j) {
            const int col = n0 + j * 16 + l15;
            // B fragment: 4x16 f32 -> 2 VGPRs (rows kb, kb+1 per half-wave)
            v2f b;
            b.x = Wqk[(size_t)kb * NDIM + col];
            b.y = Wqk[(size_t)(kb + 1) * NDIM + col];
            acc[j] = __builtin_amdgcn_wmma_f32_16x16x4_f32(
                false, a, false, b, (short)0, acc[j], false, false);
        }
    }

    // Epilogue: bias add, scale q half, store per C/D VGPR layout:
    // element v of lane l -> row m0 + v + 8*(l>=16), col = n0 + j*16 + (l&15)
#pragma unroll
    for (int j = 0; j < 8; ++j) {
        const int col = n0 + j * 16 + l15;
        const float bv = bias[col];
        const float s  = (col < CDIM) ? Q_SCALE : 1.0f;
#pragma unroll
        for (int v = 0; v < 8; ++v) {
            const int row = m0 + v + lhalf * 8;
            qk[(size_t)row * NDIM + col] = (acc[j][v] + bv) * s;
        }
    }
}

// ---------------------------------------------------------------------------
// Phase 2: superpixel-gated 7x7 neighborhood attention.
// Block = one (b, head, 16x16 pixel tile). Stage the 22x22 k-neighborhood in
// LDS transposed as [f_chunk][pixel] float4 (conflict-free b128 reads), plus
// superpixel labels. Each thread: one pixel, 49 masked dot products.
// ---------------------------------------------------------------------------
#define NB 22                 // neighborhood tile edge
#define NPIX (NB * NB)        // 484

__global__ __launch_bounds__(256) void sna_attn_kernel(
    const float* __restrict__ qk,   // [M_TOTAL, 256]; q pre-scaled in cols<128
    const int*   __restrict__ sp,   // [B, H, W]
    float*       __restrict__ out)  // [B, HEADS, H, W, 49]
{
    __shared__ float4 ktile[8 * NPIX];   // 61952 B
    __shared__ int    sptile[NPIX];      //  1936 B

    const int tid  = threadIdx.x;
    const int x0   = blockIdx.x * 16;
    const int y0   = blockIdx.y * 16;
    const int z    = blockIdx.z;          // b*HEADS + head
    const int b    = z >> 2;
    const int head = z & 3;

    const int r_lo = iclamp(y0 - 3, 0, HDIM - KS);
    const int c_lo = iclamp(x0 - 3, 0, WDIM - KS);

    // ---- stage neighborhood k-vectors + labels ----
    for (int p = tid; p < NPIX; p += 256) {
        const int py = p / NB, px = p % NB;
        const int gy = imin(r_lo + py, HDIM - 1);
        const int gx = imin(c_lo + px, WDIM - 1);
        const size_t row = (size_t)b * (HDIM * WDIM) + (size_t)gy * WDIM + gx;
        const float4* src =
            (const float4*)(qk + row * NDIM + CDIM + head * HDSZ);
#pragma unroll
        for (int f = 0; f < 8; ++f) ktile[f * NPIX + p] = src[f];
        sptile[p] = sp[row];
    }
    __syncthreads();

    // ---- per-pixel compute ----
    const int x = x0 + (tid & 15);
    const int y = y0 + (tid >> 4);
    const size_t row = (size_t)b * (HDIM * WDIM) + (size_t)y * WDIM + x;

    float4 q[8];
    const float4* qsrc = (const float4*)(qk + row * NDIM + head * HDSZ);
#pragma unroll
    for (int f = 0; f < 8; ++f) q[f] = qsrc[f];

    const int lry0 = iclamp(y - 3, 0, HDIM - KS) - r_lo;  // in [0, 15]
    const int lrx0 = iclamp(x - 3, 0, WDIM - KS) - c_lo;
    const int spc  = sptile[(y - r_lo) * NB + (x - c_lo)];

    float* op = out + (((size_t)z * HDIM + y) * WDIM + x) * (KS * KS);
    const float NEG = -__builtin_inff();

    for (int ki = 0; ki < KS; ++ki) {
        const int rbase = (lry0 + ki) * NB + lrx0;
#pragma unroll
        for (int kj = 0; kj < KS; ++kj) {
            const int pidx = rbase + kj;
            float ax = 0.f, ay = 0.f, az = 0.f, aw = 0.f;
#pragma unroll
            for (int f = 0; f < 8; ++f) {
                const float4 kf = ktile[f * NPIX + pidx];
                ax = fmaf(q[f].x, kf.x, ax);
                ay = fmaf(q[f].y, kf.y, ay);
                az = fmaf(q[f].z, kf.z, az);
                aw = fmaf(q[f].w, kf.w, aw);
            }
            const float dot = (ax + ay) + (az + aw);
            op[ki * KS + kj] = (sptile[pidx] == spc) ? dot : NEG;
        }
    }
}

// ---------------------------------------------------------------------------
extern "C" void kernel_launch(void* const* d_in, const int* in_sizes, int n_in,
                              void* d_out, int out_size, void* d_ws, size_t ws_size,
                              hipStream_t stream) {
    const float* x     = (const float*)d_in[0];
    const int*   imgSp = (const int*)  d_in[1];
    const float* Wqk   = (const float*)d_in[2];
    const float* bqk   = (const float*)d_in[3];
    float* out = (float*)d_out;
    float* qk  = (float*)d_ws;   // 262144 x 256 f32 = 256 MiB scratch

    // Phase 1: 16384 M tiles x 2 N groups = 32768 wave tasks, 8 waves/block
    qk_gemm_wmma<<<dim3(4096), dim3(256), 0, stream>>>(x, Wqk, bqk, qk);

    // Phase 2: 16x16 pixel tiles, z = b*HEADS + head
    sna_attn_kernel<<<dim3(WDIM / 16, HDIM / 16, BATCH * HEADS),
                      dim3(256), 0, stream>>>(qk, imgSp, out);
}